// DotPredictor_12773232738509
// MI455X (gfx1250) — compile-verified
//
#include <hip/hip_runtime.h>
#include <hip/hip_bf16.h>

// MI455X / gfx1250 edge-dot kernel.
// score[e] = dot(h[u[e]], h[v[e]]), D=128, fp32.
// Memory-bound (0.25 FLOP/B); h (51 MB) is L2-resident (192 MB L2).
// Strategy: coalesced 512B row gathers -> LDS, then chain 32x
// V_WMMA_F32_16X16X4_F32 over K=128 for 16 edges per wave; the diagonal of
// the 16x16 fp32 accumulator is the 16 edge scores.

typedef float v2f __attribute__((ext_vector_type(2)));
typedef float v4f __attribute__((ext_vector_type(4)));
typedef float v8f __attribute__((ext_vector_type(8)));

#define D_FEAT      128
#define ROWSTRIDE   132   // 128 + 4 pad: bank = (4*row + col) % 64 -> conflict-free b64 frag reads
#define TILE        16    // edges per wave (one WMMA N/M extent)
#define WAVES_PB    8     // waves per block (256 threads, wave32)

__global__ __launch_bounds__(256)
void edge_dot_wmma(const float* __restrict__ h,
                   const int*   __restrict__ u,
                   const int*   __restrict__ v,
                   float*       __restrict__ out,
                   int nE)
{
    extern __shared__ float smem[];

    const int  lane   = threadIdx.x & 31;
    const int  wave   = threadIdx.x >> 5;
    const long nTiles = ((long)nE + TILE - 1) / TILE;
    const long tileId = (long)blockIdx.x * WAVES_PB + wave;
    const bool active = (tileId < nTiles);
    const long ebase  = active ? tileId * TILE : 0;

    // Per-wave private LDS slices: A = u-rows, B = v-rows (16 x 132 floats each).
    float* ldsA = smem + (size_t)wave * (2 * TILE * ROWSTRIDE);
    float* ldsB = ldsA + TILE * ROWSTRIDE;

    // ---- Phase 1: coalesced gather. One full 512B row per wave-wide b128 load.
    #pragma unroll 4
    for (int r = 0; r < TILE; ++r) {
        const long e  = ebase + r;
        int ur = 0, vr = 0;
        if (e < (long)nE) { ur = u[e]; vr = v[e]; }          // uniform -> scalar loads
        const v4f au = *(const v4f*)(h + (size_t)ur * D_FEAT + lane * 4);
        const v4f bv = *(const v4f*)(h + (size_t)vr * D_FEAT + lane * 4);
        *(v4f*)(ldsA + r * ROWSTRIDE + lane * 4) = au;
        *(v4f*)(ldsB + r * ROWSTRIDE + lane * 4) = bv;
    }
    __syncthreads();   // make LDS writes visible across lanes of the wave

    // ---- Phase 2: 32 chained V_WMMA_F32_16X16X4_F32 over K = 0..127.
    // A frag layout (16x4 f32): lane L holds M = L&15; VGPR j holds K = 2*(L>>4) + j.
    // B frag uses the mirrored layout with N = L&15. Both are float2 LDS reads.
    const int    mrow  = lane & 15;
    const int    khalf = (lane >> 4) * 2;
    const float* Arow  = ldsA + mrow * ROWSTRIDE + khalf;
    const float* Brow  = ldsB + mrow * ROWSTRIDE + khalf;

    v8f c = {};   // 16x16 fp32 accumulator (8 VGPRs)
    #pragma unroll
    for (int k = 0; k < D_FEAT; k += 4) {
        const v2f a = *(const v2f*)(Arow + k);
        const v2f b = *(const v2f*)(Brow + k);
        // (neg_a, A, neg_b, B, c_mod, C, reuse_a, reuse_b)
        c = __builtin_amdgcn_wmma_f32_16x16x4_f32(false, a, false, b,
                                                  (short)0, c, false, false);
    }

    // ---- Phase 3: extract diagonal D[m][m].
    // C/D layout: VGPR r -> lanes 0-15: (M=r, N=lane); lanes 16-31: (M=r+8, N=lane-16).
    // Diagonal lives in lanes 0-7 (m=lane, vgpr=lane) and lanes 24-31 (m=lane-16, vgpr=lane-24).
    const bool validLane = (lane < 8) || (lane >= 24);
    const int  m   = (lane < 8) ? lane : (lane - 16);
    const int  sel = (lane < 8) ? lane : (lane - 24);

    float val = c[0];
    #pragma unroll
    for (int i = 1; i < 8; ++i)
        val = (sel == i) ? c[i] : val;   // 7x v_cndmask select chain

    const long e = ebase + m;
    if (active && validLane && e < (long)nE)
        out[e] = val;
}

extern "C" void kernel_launch(void* const* d_in, const int* in_sizes, int n_in,
                              void* d_out, int out_size, void* d_ws, size_t ws_size,
                              hipStream_t stream)
{
    const float* h  = (const float*)d_in[0];
    const int*   u  = (const int*)d_in[1];   // harness delivers integer inputs as int32
    const int*   v  = (const int*)d_in[2];
    float*       out = (float*)d_out;

    const int  nE     = in_sizes[1];
    const long nTiles = ((long)nE + TILE - 1) / TILE;
    const int  blocks = (int)((nTiles + WAVES_PB - 1) / WAVES_PB);
    const size_t smemBytes = (size_t)WAVES_PB * 2 * TILE * ROWSTRIDE * sizeof(float); // 132 KB

    edge_dot_wmma<<<blocks, 256, smemBytes, stream>>>(h, u, v, out, nE);
}